// SpikingIrisClassifier_55671366090812
// MI455X (gfx1250) — compile-verified
//
#include <hip/hip_runtime.h>

// ---- Problem constants (match reference) ----
#define BETA    0.9f
#define THRESH  1.0f
#define T_STEPS 50
#define BATCH   262144
#define IN_F    4
#define HID     16
#define OUT_F   3

typedef float v2f __attribute__((ext_vector_type(2)));
typedef float v8f __attribute__((ext_vector_type(8)));

// SWAPX16: ds_swizzle group-of-32, xor_mask=0x10, or=0, and=0x1f -> offset 0x401F
__device__ __forceinline__ float swap_x16(float v) {
    int i = __builtin_amdgcn_ds_swizzle(__float_as_int(v), 0x401F);
    return __int_as_float(i);
}

// One wave32 handles a tile of 16 batch elements for all 50 timesteps.
//   D(16x16) = W1(16x4) x x_t^T(4x16) + (BETA*mem1 - spk1_prev)   [V_WMMA_F32_16X16X4_F32]
// C/D layout: lane = batch col (lane&15), VGPR r = hidden row r + 8*(lane>>4).
__global__ __launch_bounds__(256) void SpikingIris_snn_kernel(
    const float* __restrict__ x,    // [T, BATCH, IN_F]
    const float* __restrict__ W1,   // [HID, IN_F]
    const float* __restrict__ W2,   // [OUT_F, HID]
    float* __restrict__ out)        // [BATCH, OUT_F]
{
    const int lane = threadIdx.x & 31;
    const int half = lane >> 4;          // 0: lanes 0-15, 1: lanes 16-31
    const int m    = lane & 15;          // A row / batch column index
    const int wave = threadIdx.x >> 5;
    const int tile = blockIdx.x * (blockDim.x >> 5) + wave;
    const int b0   = tile << 4;          // first batch element of this tile

    // ---- A = W1 (16x4 f32), loop-invariant. ISA A layout:
    //   lanes 0-15:  VGPR0=K0, VGPR1=K1 ; lanes 16-31: VGPR0=K2, VGPR1=K3
    v2f amat;
    amat.x = W1[m * IN_F + 2 * half + 0];
    amat.y = W1[m * IN_F + 2 * half + 1];

    // ---- Per-lane W2 slice: this lane owns hidden rows (8*half .. 8*half+7)
    float w2s[OUT_F][8];
#pragma unroll
    for (int o = 0; o < OUT_F; ++o)
#pragma unroll
        for (int r = 0; r < 8; ++r)
            w2s[o][r] = W2[o * HID + 8 * half + r];

    // ---- State ----
    v8f mem1;
    float spk1[8];
#pragma unroll
    for (int r = 0; r < 8; ++r) { mem1[r] = 0.0f; spk1[r] = 0.0f; }
    float mem2_0 = 0.f, mem2_1 = 0.f, mem2_2 = 0.f;
    float spk2_0 = 0.f, spk2_1 = 0.f, spk2_2 = 0.f;   // also serves as next reset
    float acc0 = 0.f, acc1 = 0.f, acc2 = 0.f;

    // Per-lane element offset into x for timestep 0 (64-bit safe; 8B aligned)
    const size_t xoff = (size_t)(b0 + m) * IN_F + (size_t)(2 * half);
    const size_t tstride = (size_t)BATCH * IN_F;

#pragma unroll 2
    for (int t = 0; t < T_STEPS; ++t) {
        const float* px = x + (size_t)t * tstride + xoff;
        // B = x_t^T (4x16): one b64 load -> (K = 2*half, 2*half+1), N = m
        v2f bmat = *(const v2f*)px;
        if (t + 4 < T_STEPS)  // uniform branch: EXEC untouched
            __builtin_prefetch(px + 4 * tstride, 0, 1);  // global_prefetch_b8

        // Fold LIF-1 decay+reset into the accumulator: C = BETA*mem1 - spk1_prev
        v8f cacc;
#pragma unroll
        for (int r = 0; r < 8; ++r)
            cacc[r] = __builtin_fmaf(BETA, mem1[r], -spk1[r]);  // THRESH==1

        // mem1_new = W1 @ x_t^T + C   (exact fp32, K=4)
        mem1 = __builtin_amdgcn_wmma_f32_16x16x4_f32(
            false, amat, false, bmat, (short)0, cacc, false, false);

        // spk1 + binary-weighted layer-2 partials (hidden rows owned by this lane)
        float p0 = 0.f, p1 = 0.f, p2 = 0.f;
#pragma unroll
        for (int r = 0; r < 8; ++r) {
            float s = (mem1[r] > THRESH) ? 1.0f : 0.0f;
            spk1[r] = s;
            p0 = __builtin_fmaf(s, w2s[0][r], p0);
            p1 = __builtin_fmaf(s, w2s[1][r], p1);
            p2 = __builtin_fmaf(s, w2s[2][r], p2);
        }

        // Combine the two hidden halves across lane pairs (L <-> L^16)
        float c0 = p0 + swap_x16(p0);
        float c1 = p1 + swap_x16(p1);
        float c2 = p2 + swap_x16(p2);

        // LIF-2: mem2 = BETA*mem2 + cur2 - reset ; reset == spk2_prev
        mem2_0 = __builtin_fmaf(BETA, mem2_0, c0 - spk2_0);
        mem2_1 = __builtin_fmaf(BETA, mem2_1, c1 - spk2_1);
        mem2_2 = __builtin_fmaf(BETA, mem2_2, c2 - spk2_2);
        spk2_0 = (mem2_0 > THRESH) ? 1.0f : 0.0f;
        spk2_1 = (mem2_1 > THRESH) ? 1.0f : 0.0f;
        spk2_2 = (mem2_2 > THRESH) ? 1.0f : 0.0f;
        acc0 += spk2_0; acc1 += spk2_1; acc2 += spk2_2;
    }

    // Lanes 0-15 hold the full (redundant) result for batch b0+m
    if (half == 0) {
        float* po = out + (size_t)(b0 + m) * OUT_F;
        po[0] = acc0; po[1] = acc1; po[2] = acc2;
    }
}

extern "C" void kernel_launch(void* const* d_in, const int* in_sizes, int n_in,
                              void* d_out, int out_size, void* d_ws, size_t ws_size,
                              hipStream_t stream) {
    (void)in_sizes; (void)n_in; (void)d_ws; (void)ws_size; (void)out_size;
    const float* x  = (const float*)d_in[0];   // [50, 262144, 4] f32
    const float* W1 = (const float*)d_in[1];   // [16, 4] f32
    const float* W2 = (const float*)d_in[2];   // [3, 16] f32
    float* out = (float*)d_out;                // [262144, 3] f32

    // 16 batch / wave, 8 waves / block -> 128 batch / block
    const int tiles  = BATCH / 16;             // 16384 waves
    const int blocks = tiles / 8;              // 2048 blocks of 256 threads
    hipLaunchKernelGGL(SpikingIris_snn_kernel, dim3(blocks), dim3(256), 0, stream,
                       x, W1, W2, out);
}